// HashGridEncode_1614907703695
// MI455X (gfx1250) — compile-verified
//
#include <hip/hip_runtime.h>
#include <math.h>

// Instant-NGP hash grid config (compile-time)
#define NLEVELS 16
#define LEVEL_SIZE 524288u      // 2^19 entries per level
#define LEVEL_MASK 524287u      // h % LEVEL_SIZE == h & LEVEL_MASK
#define PRIME_Y 2654435761u
#define PRIME_Z 805459861u

// Compile-time resolutions (constant-folded after full unroll)
constexpr int kRes[NLEVELS] = {
    16, 22, 30, 42, 58, 80, 111, 154, 212, 294, 406, 561, 776, 1072, 1482, 2048};

struct LevelSetup {
    unsigned hx0, hx1, hy0, hy1, hz0, hz1;   // incremental hash terms (prime_x==1)
    float wx0, wx1;                          // x lerp weights
    float wyz00, wyz10, wyz01, wyz11;        // shared y*z weight products
};

__device__ __forceinline__ LevelSetup level_setup(float nx, float ny, float nz, int r) {
    const float scale = (float)(r - 1);
    const float px = nx * scale;
    const float py = ny * scale;
    const float pz = nz * scale;

    // trunc == floor for pos >= 0 (all valid points); invalid points are masked.
    int fx = (int)px;
    int fy = (int)py;
    int fz = (int)pz;
    fx = fx < 0 ? 0 : (fx > r - 2 ? r - 2 : fx);
    fy = fy < 0 ? 0 : (fy > r - 2 ? r - 2 : fy);
    fz = fz < 0 ? 0 : (fz > r - 2 ? r - 2 : fz);

    // frac from the clipped base, matching the reference.
    const float tx = px - (float)fx;
    const float ty = py - (float)fy;
    const float tz = pz - (float)fz;

    LevelSetup s;
    s.hx0 = (unsigned)fx;            s.hx1 = s.hx0 + 1u;
    s.hy0 = (unsigned)fy * PRIME_Y;  s.hy1 = s.hy0 + PRIME_Y;
    s.hz0 = (unsigned)fz * PRIME_Z;  s.hz1 = s.hz0 + PRIME_Z;

    const float wy0 = 1.0f - ty, wy1 = ty;
    const float wz0 = 1.0f - tz, wz1 = tz;
    s.wx0 = 1.0f - tx;  s.wx1 = tx;
    s.wyz00 = wy0 * wz0;
    s.wyz10 = wy1 * wz0;
    s.wyz01 = wy0 * wz1;
    s.wyz11 = wy1 * wz1;
    return s;
}

__device__ __forceinline__ unsigned corner_idx(const LevelSetup& s, int c) {
    const unsigned h = ((c & 1) ? s.hx1 : s.hx0) ^
                       ((c & 2) ? s.hy1 : s.hy0) ^
                       ((c & 4) ? s.hz1 : s.hz0);
    return h & LEVEL_MASK;
}

__device__ __forceinline__ float corner_w(const LevelSetup& s, int c) {
    const float wyz = (c & 4) ? ((c & 2) ? s.wyz11 : s.wyz01)
                              : ((c & 2) ? s.wyz10 : s.wyz00);
    return ((c & 1) ? s.wx1 : s.wx0) * wyz;
}

__global__ __launch_bounds__(256) void hashgrid_encode_kernel(
    const float* __restrict__ xyz,        // (B,3)
    const float* __restrict__ emb,        // (16*524288, 2)
    const float* __restrict__ mn,         // (3,)
    const float* __restrict__ mx,         // (3,)
    float* __restrict__ out,              // (B, 32)
    int B)
{
    const int i = blockIdx.x * blockDim.x + threadIdx.x;
    if (i >= B) return;

    // Warm L2 for coordinates one grid-stride ahead (gfx1250 global_prefetch_b8).
    const int stride = gridDim.x * blockDim.x;
    if (i + stride < B) {
        __builtin_prefetch(xyz + 3 * (size_t)(i + stride), 0, 1);
    }

    const float x = xyz[3 * (size_t)i + 0];
    const float y = xyz[3 * (size_t)i + 1];
    const float z = xyz[3 * (size_t)i + 2];
    const float mnx = mn[0], mny = mn[1], mnz = mn[2];
    const float nx = (x - mnx) / (mx[0] - mnx);
    const float ny = (y - mny) / (mx[1] - mny);
    const float nz = (z - mnz) / (mx[2] - mnz);

    const bool valid = (nx >= 0.0f) & (nx <= 1.0f) &
                       (ny >= 0.0f) & (ny <= 1.0f) &
                       (nz >= 0.0f) & (nz <= 1.0f);

    float4* __restrict__ orow = (float4*)(out + (size_t)i * (2 * NLEVELS));

    // Two levels per phase: 16 independent b64 gathers in flight, result is
    // exactly one float4 -> store immediately (no long-lived accumulator file).
#pragma unroll
    for (int p = 0; p < NLEVELS / 2; ++p) {
        const int lA = 2 * p, lB = 2 * p + 1;
        const LevelSetup A = level_setup(nx, ny, nz, kRes[lA]);
        const LevelSetup B2 = level_setup(nx, ny, nz, kRes[lB]);

        const float2* __restrict__ baseA = (const float2*)emb + (size_t)lA * LEVEL_SIZE;
        const float2* __restrict__ baseB = (const float2*)emb + (size_t)lB * LEVEL_SIZE;

        float2 fA[8], fB[8];
#pragma unroll
        for (int c = 0; c < 8; ++c) fA[c] = baseA[corner_idx(A, c)];
#pragma unroll
        for (int c = 0; c < 8; ++c) fB[c] = baseB[corner_idx(B2, c)];

        float s0A = 0.0f, s1A = 0.0f, s0B = 0.0f, s1B = 0.0f;
#pragma unroll
        for (int c = 0; c < 8; ++c) {
            const float wA = corner_w(A, c);
            s0A = fmaf(wA, fA[c].x, s0A);
            s1A = fmaf(wA, fA[c].y, s1A);
        }
#pragma unroll
        for (int c = 0; c < 8; ++c) {
            const float wB = corner_w(B2, c);
            s0B = fmaf(wB, fB[c].x, s0B);
            s1B = fmaf(wB, fB[c].y, s1B);
        }

        orow[p] = make_float4(valid ? s0A : 0.0f,
                              valid ? s1A : 0.0f,
                              valid ? s0B : 0.0f,
                              valid ? s1B : 0.0f);
    }
}

extern "C" void kernel_launch(void* const* d_in, const int* in_sizes, int n_in,
                              void* d_out, int out_size, void* d_ws, size_t ws_size,
                              hipStream_t stream) {
    const float* xyz = (const float*)d_in[0];
    const float* emb = (const float*)d_in[1];
    const float* mn  = (const float*)d_in[2];
    const float* mx  = (const float*)d_in[3];
    float* out = (float*)d_out;

    const int B = in_sizes[0] / 3;
    const int threads = 256;            // 8 wave32 waves per block
    const int blocks = (B + threads - 1) / threads;
    hipLaunchKernelGGL(hashgrid_encode_kernel, dim3(blocks), dim3(threads), 0,
                       stream, xyz, emb, mn, mx, out, B);
}